// HarmonicOscillator_19705309954212
// MI455X (gfx1250) — compile-verified
//
#include <hip/hip_runtime.h>
#include <hip/hip_bf16.h>
#include <math.h>

// Problem constants (from reference)
#define TT     600      // frames
#define NOSC   64
#define HOP    240
#define SAMPS  (TT*HOP) // 144000
#define INV_SR (1.0f/48000.0f)
#define NYQ    24000.0f
#define NCHUNK 19       // ceil(600/32)

// ---------------------------------------------------------------------------
// Kernel 1: per-(batch,frame) sum of linearly-upsampled f0 / SR (closed form)
// framesum[b,k] = (1/SR) * sum_{r=0..239} upsample(f0)[b, k*240+r]
// Clamped neighbor loads make edge frames (k==0, k==T-1) exact, no branches.
// ---------------------------------------------------------------------------
__global__ void hosc_framesum(const float* __restrict__ f0,
                              float* __restrict__ framesum, int total) {
    int idx = blockIdx.x * blockDim.x + threadIdx.x;
    if (idx >= total) return;
    int b = idx / TT;
    int k = idx - b * TT;
    const float* f = f0 + b * TT;
    int km = (k > 0) ? k - 1 : 0;
    int kp = (k < TT - 1) ? k + 1 : TT - 1;
    float xm = f[km], xc = f[k], xp = f[kp];
    // first half (r=0..119): i0=k-1, sum(w)=90 ; second half: i0=k, sum(w)=30
    float s = 120.0f * xm + 90.0f * (xc - xm)
            + 120.0f * xc + 30.0f * (xp - xc);
    framesum[idx] = s * INV_SR;
}

// ---------------------------------------------------------------------------
// Kernel 2: exclusive prefix over 600 frame sums per batch.
// One wave32 per batch: preload all chunks (pipelined loads), then 19
// register-only rounds of shfl_up inclusive scan + carried prefix.
// ---------------------------------------------------------------------------
__global__ __launch_bounds__(32)
void hosc_scan(const float* __restrict__ framesum,
               float* __restrict__ base, int B) {
    const int b = blockIdx.x;
    const int l = threadIdx.x;          // 0..31
    const float* f = framesum + b * TT;
    float* o = base + b * TT;

    float vals[NCHUNK];
#pragma unroll
    for (int r = 0; r < NCHUNK; ++r) {
        int idx = r * 32 + l;
        vals[r] = (idx < TT) ? f[idx] : 0.0f;
    }

    float carry = 0.0f;
#pragma unroll
    for (int r = 0; r < NCHUNK; ++r) {
        float incl = vals[r];
#pragma unroll
        for (int d = 1; d < 32; d <<= 1) {
            float t = __shfl_up(incl, d, 32);
            if (l >= d) incl += t;
        }
        float excl = __shfl_up(incl, 1, 32);
        if (l == 0) excl = 0.0f;
        excl += carry;
        int idx = r * 32 + l;
        if (idx < TT) o[idx] = excl;
        carry += __shfl(incl, 31, 32);
    }
}

// ---------------------------------------------------------------------------
// Kernel 3: main synthesis. One block = one (batch, frame) = 240 samples.
// Coefficient tile c[b, :, k-1..k+1] (3x64 floats) staged into LDS with
// CDNA5 async global->LDS loads (ASYNCcnt), Nyquist-masked in place, then
// 240 lanes each evaluate 64 partials with hardware v_sin_f32.
// ---------------------------------------------------------------------------
__global__ __launch_bounds__(256)
void hosc_synth(const float* __restrict__ f0,
                const float* __restrict__ a,
                const float* __restrict__ c,
                const float* __restrict__ base,
                float* __restrict__ out) {
    const int blk = blockIdx.x;
    const int b = blk / TT;
    const int k = blk - b * TT;
    const int tid = threadIdx.x;

    __shared__ float cm[3 * NOSC];   // c (masked) for frames clamp(k-1+j)

    // --- stage coefficient tile via async global->LDS (CDNA5 path) ---------
    if (tid < 3 * NOSC) {
        int j = tid >> 6;            // 0..2 -> frame slot
        int n = tid & 63;            // oscillator
        int tf = k - 1 + j;
        tf = (tf < 0) ? 0 : ((tf > TT - 1) ? TT - 1 : tf);
        const float* gp = c + ((size_t)(b * NOSC + n)) * TT + tf;
        unsigned ldsoff = (unsigned)(size_t)(&cm[j * NOSC + n]); // addr[31:0] = LDS offset
        asm volatile("global_load_async_to_lds_b32 %0, %1, off"
                     :: "v"(ldsoff), "v"(gp) : "memory");
    }

    // scalar frame values (tiny, L2-resident) overlap the async transfer
    const float* fb = f0 + b * TT;
    const int km = (k > 0) ? k - 1 : 0;
    const int kp = (k < TT - 1) ? k + 1 : TT - 1;
    const float xm = fb[km], xc = fb[k], xp = fb[kp];
    const float basev = base[b * TT + k];

    asm volatile("s_wait_asynccnt 0x0" ::: "memory");
    __syncthreads();

    // Nyquist mask applied once in LDS: (n+1)*f0[b,tf] < 24000 ? keep : 0
    if (tid < 3 * NOSC) {
        int j = tid >> 6;
        int n = tid & 63;
        int tf = k - 1 + j;
        tf = (tf < 0) ? 0 : ((tf > TT - 1) ? TT - 1 : tf);
        float fn = (float)(n + 1) * fb[tf];
        if (!(fn < NYQ)) cm[j * NOSC + n] = 0.0f;
    }
    __syncthreads();

    if (tid >= HOP) return;
    const int r = tid;
    const int jj = k * HOP + r;

    // --- phase: base + closed-form inclusive within-frame partial ----------
    float partial;
    if (r < 120) {
        float R = (float)(r + 1);
        partial = R * xm + (xc - xm) * (R * (0.5f + R * (1.0f / 480.0f)));
    } else {
        float M = (float)(r - 120 + 1);
        partial = 120.0f * xm + 90.0f * (xc - xm)
                + M * xc + (xp - xc) * (M * M * (1.0f / 480.0f));
    }
    const float phi = basev + partial * INV_SR;   // phase in cycles

    // --- interpolation weight, exactly as reference ------------------------
    float real = ((float)jj + 0.5f) / 240.0f - 0.5f;
    real = fmaxf(real, 0.0f);
    float fi0 = floorf(real);
    float w = real - fi0;
    int i0 = (int)fi0;
    int i1 = (i0 + 1 < TT - 1) ? i0 + 1 : TT - 1;
    float omw = 1.0f - w;

    const float* ab = a + b * TT;
    const float loud = ab[i0] * omw + ab[i1] * w;

    // map absolute frame indices to LDS slots (slot j holds clamp(k-1+j))
    const int sa = i0 - (k - 1);   // in {0,1}
    const int sb = i1 - (k - 1);   // in {1,2}
    const float2* ca = (const float2*)(&cm[sa * NOSC]);
    const float2* cb = (const float2*)(&cm[sb * NOSC]);

    // --- 64-partial oscillator bank: v_sin_f32 + FMA, ds_load_b64 pairs ---
    float sum = 0.0f;
#pragma unroll
    for (int h = 0; h < NOSC / 2; ++h) {
        float2 va = ca[h];
        float2 vb = cb[h];
        float n1 = (float)(2 * h + 1);
        float n2 = (float)(2 * h + 2);
        float t1 = n1 * phi;  t1 = t1 - rintf(t1);   // exact fractional cycles
        float t2 = n2 * phi;  t2 = t2 - rintf(t2);
        float s1 = __builtin_amdgcn_sinf(t1);        // v_sin_f32: sin(2*pi*x)
        float s2 = __builtin_amdgcn_sinf(t2);
        float amp1 = va.x * omw + vb.x * w;
        float amp2 = va.y * omw + vb.y * w;
        sum = fmaf(s1, amp1, sum);
        sum = fmaf(s2, amp2, sum);
    }

    out[(size_t)b * SAMPS + jj] = sum * 0.02f * loud;
}

// ---------------------------------------------------------------------------
extern "C" void kernel_launch(void* const* d_in, const int* in_sizes, int n_in,
                              void* d_out, int out_size, void* d_ws, size_t ws_size,
                              hipStream_t stream) {
    const float* f0 = (const float*)d_in[0];   // [B, 600]
    const float* a  = (const float*)d_in[1];   // [B, 600]
    const float* c  = (const float*)d_in[2];   // [B, 64, 600]
    float* out = (float*)d_out;                // [B, 144000]

    const int B = in_sizes[0] / TT;            // 8
    const int BF = B * TT;                     // 4800 frames total

    float* framesum = (float*)d_ws;            // BF floats
    float* basep    = framesum + BF;           // BF floats

    hosc_framesum<<<(BF + 255) / 256, 256, 0, stream>>>(f0, framesum, BF);
    hosc_scan<<<B, 32, 0, stream>>>(framesum, basep, B);
    hosc_synth<<<BF, 256, 0, stream>>>(f0, a, c, basep, out);
}